// BoundaryLoss_44762149159187
// MI455X (gfx1250) — compile-verified
//
#include <hip/hip_runtime.h>
#include <hip/hip_bf16.h>

typedef __attribute__((ext_vector_type(2))) float v2f;
typedef __attribute__((ext_vector_type(8))) float v8f;

#define BIGV 1.0e6f

// ---------------------------------------------------------------------------
// Kernel A: 1-D L1 distance scans along H for both masks (target==1, target==0),
// then square. One thread per (b, w) column; consecutive threads are consecutive
// w -> every global access in the h-loop is a fully coalesced 1 KB row segment.
// Writes G_out = g1^2 and G_in = g1^2 into workspace.
// ---------------------------------------------------------------------------
__global__ void edt_col_kernel(const int* __restrict__ target,
                               float* __restrict__ Gout,
                               float* __restrict__ Gin,
                               int B, int H, int W) {
    int idx = blockIdx.x * blockDim.x + threadIdx.x;
    if (idx >= B * W) return;
    int b = idx / W;
    int w = idx - b * W;
    size_t base = (size_t)b * H * W + w;
    const int* t = target + base;
    float* go = Gout + base;
    float* gi = Gin + base;

    // forward scan: c = min(row, c+1), c0 = BIG   (matches jax.lax.scan step)
    float co = BIGV, ci = BIGV;
    for (int h = 0; h < H; ++h) {
        int tv = t[(size_t)h * W];
        float gvo = (tv == 1) ? 0.0f : BIGV;
        float gvi = (tv == 0) ? 0.0f : BIGV;
        co = fminf(gvo, co + 1.0f);
        ci = fminf(gvi, ci + 1.0f);
        go[(size_t)h * W] = co;
        gi[(size_t)h * W] = ci;
    }
    // backward scan, combine with forward, square
    co = BIGV; ci = BIGV;
    for (int h = H - 1; h >= 0; --h) {
        int tv = t[(size_t)h * W];
        float gvo = (tv == 1) ? 0.0f : BIGV;
        float gvi = (tv == 0) ? 0.0f : BIGV;
        co = fminf(gvo, co + 1.0f);
        ci = fminf(gvi, ci + 1.0f);
        float g1o = fminf(go[(size_t)h * W], co);
        float g1i = fminf(gi[(size_t)h * W], ci);
        go[(size_t)h * W] = g1o * g1o;   // BIG^2 = 1e12, finite in fp32
        gi[(size_t)h * W] = g1i * g1i;
    }
}

// ---------------------------------------------------------------------------
// Kernel B: exact lower envelope along W + fused sigmoid*dist partial loss.
// One block per (b,h) row, 256 threads; both G rows cached in LDS, each thread
// does the 256-long min-plus for its output column for BOTH transforms at once
// (shared (i-k)^2 term). Deterministic fixed-tree block reduction -> rowSums.
// ---------------------------------------------------------------------------
__global__ void edt_row_loss_kernel(const float* __restrict__ pred,
                                    const float* __restrict__ Gout,
                                    const float* __restrict__ Gin,
                                    float* __restrict__ rowSums,
                                    int W) {
    __shared__ float sGo[256];
    __shared__ float sGi[256];
    __shared__ float sRed[256];

    int row = blockIdx.x;            // b*H + h
    int i = threadIdx.x;
    size_t off = (size_t)row * W;

    sGo[i] = Gout[off + i];
    sGi[i] = Gin[off + i];
    __syncthreads();

    float fi = (float)i;
    float d2o = 3.0e38f, d2i = 3.0e38f;
#pragma unroll 4
    for (int k = 0; k < 256; ++k) {
        float dk = fi - (float)k;
        float t = dk * dk;
        d2o = fminf(d2o, sGo[k] + t);
        d2i = fminf(d2i, sGi[k] + t);
    }
    float dist = sqrtf(d2o) + sqrtf(d2i);
    float p = pred[off + i];
    float prob = 1.0f / (1.0f + __expf(-p));
    sRed[i] = prob * dist;
    __syncthreads();

    for (int s = 128; s > 0; s >>= 1) {
        if (i < s) sRed[i] += sRed[i + s];
        __syncthreads();
    }
    if (i == 0) rowSums[row] = sRed[0];
}

// ---------------------------------------------------------------------------
// Kernel C: reduce nRows (=2048, multiple of 64) row sums with
// V_WMMA_F32_16X16X4_F32:  C += A(16x4 data chunk) x B(4x16 ones).
// After the loop D[m][n] = total of A-row m over all chunks (same for all n);
// column 0 of D lives in lanes 0 (M=0..7) and 16 (M=8..15), 8 VGPRs each.
// Single wave (32 threads), EXEC all ones as WMMA requires.
// ---------------------------------------------------------------------------
__global__ void wmma_reduce_kernel(const float* __restrict__ rowSums,
                                   float* __restrict__ out,
                                   int n, float scale) {
    int lane = threadIdx.x;          // 0..31
    int m = lane & 15;
    int k0 = (lane >= 16) ? 2 : 0;

    v8f acc = {};
    v2f bones;
    bones.x = 1.0f; bones.y = 1.0f;

    for (int base = 0; base < n; base += 64) {
        const float* chunk = rowSums + base;     // 16 rows x 4 k-values
        v2f a;
        a.x = chunk[m * 4 + k0];
        a.y = chunk[m * 4 + k0 + 1];
        acc = __builtin_amdgcn_wmma_f32_16x16x4_f32(
            /*neg_a=*/false, a, /*neg_b=*/false, bones,
            /*c_mod=*/(short)0, acc, /*reuse_a=*/false, /*reuse_b=*/false);
    }

    // per-lane: sum of this lane's 8 D rows (valid columns: lane0 -> N=0 rows 0..7,
    // lane16 -> N=0 rows 8..15)
    float s = acc[0] + acc[1] + acc[2] + acc[3] +
              acc[4] + acc[5] + acc[6] + acc[7];
    float total = __shfl(s, 0, 32) + __shfl(s, 16, 32);
    if (lane == 0) out[0] = total * scale;
}

// ---------------------------------------------------------------------------
extern "C" void kernel_launch(void* const* d_in, const int* in_sizes, int n_in,
                              void* d_out, int out_size, void* d_ws, size_t ws_size,
                              hipStream_t stream) {
    const float* pred = (const float*)d_in[0];
    const int* target = (const int*)d_in[1];
    float* out = (float*)d_out;

    const int H = 256, W = 256;
    int n = in_sizes[0];                 // B*H*W = 524288
    int B = n / (H * W);                 // 8
    int rows = B * H;                    // 2048 (multiple of 64)
    int cols = B * W;                    // 2048

    float* Gout = (float*)d_ws;          // n floats
    float* Gin = Gout + n;               // n floats
    float* rowSums = Gin + n;            // rows floats   (total ~4.2 MB)

    edt_col_kernel<<<(cols + 255) / 256, 256, 0, stream>>>(target, Gout, Gin, B, H, W);
    edt_row_loss_kernel<<<rows, 256, 0, stream>>>(pred, Gout, Gin, rowSums, W);

    float scale = 0.5f / (float)n;       // BETA / (B*H*W)
    wmma_reduce_kernel<<<1, 32, 0, stream>>>(rowSums, out, rows, scale);
}